// RGAT_Net_Graph_71975061946538
// MI455X (gfx1250) — compile-verified
//
#include <hip/hip_runtime.h>
#include <cstdint>
#include <cstddef>

// ---------------------------------------------------------------------------
// RGAT for MI455X (gfx1250, wave32). fp32 end-to-end, dense parts on
// V_WMMA_F32_16X16X4_F32. Attention restructured so the [8,N,64] per-relation
// tensor is never materialized (qr/kr projection trick + per-relation
// transform-then-scatter), keeping the working set ~45 MB and the problem
// limited by the unavoidable 800K-edge gathers at 23.3 TB/s HBM.
// ---------------------------------------------------------------------------

typedef __attribute__((ext_vector_type(2))) float v2f;
typedef __attribute__((ext_vector_type(8))) float v8f;

#define HID    64
#define OUTC   32
#define NREL   8
#define NGRAPH 64
#define NEG_SLOPE 0.2f

// ------------------------- helpers -----------------------------------------

__device__ __forceinline__ void atomicMaxF(float* addr, float val) {
  // standard monotonic-bits trick; m initialized to -inf
  if (val >= 0.0f) {
    atomicMax((int*)addr, __float_as_int(val));
  } else {
    atomicMin((unsigned int*)addr, __float_as_uint(val));
  }
}

// ------------------------- WMMA GEMM  C[M,64] = A[M,K] @ B[K,64] (+bias) ----
// block = 128 threads (4 wave32); wave w -> 16x16 tile at cols [16w,16w+16)
// grid.x = M/16 row tiles. fp32 WMMA: A frag = v2f, B frag = v2f, C = v8f.
__global__ void wmma_gemm_n64(const float* __restrict__ A,
                              const float* __restrict__ B,
                              const float* __restrict__ bias,
                              float* __restrict__ C,
                              int M, int K) {
  const int lane  = threadIdx.x & 31;
  const int wave  = threadIdx.x >> 5;          // 0..3 -> column tile
  const int row0  = blockIdx.x << 4;
  const int col0  = wave << 4;
  const int mrow  = lane & 15;                 // A: M = lane&15
  const int khalf = (lane >> 4) << 1;          // A/B: K sub-offset 0 or 2
  const int col   = col0 + (lane & 15);

  int r = row0 + mrow;
  if (r > M - 1) r = M - 1;                    // clamp (M%16==0 in practice)
  const float* Arow = A + (size_t)r * K;

  v8f acc = {0.f, 0.f, 0.f, 0.f, 0.f, 0.f, 0.f, 0.f};

  for (int k0 = 0; k0 < K; k0 += 4) {
    v2f a, b;
    // 32-bit A 16x4 layout: v0 = K (khalf), v1 = K (khalf+1)
    a.x = Arow[k0 + khalf];
    a.y = Arow[k0 + khalf + 1];
    // 32-bit B 4x16 layout (row striped across lanes, mirrored K split)
    b.x = B[(size_t)(k0 + khalf) * HID + col];
    b.y = B[(size_t)(k0 + khalf + 1) * HID + col];
    acc = __builtin_amdgcn_wmma_f32_16x16x4_f32(
        /*neg_a=*/false, a, /*neg_b=*/false, b,
        /*c_mod=*/(short)0, acc, /*reuse_a=*/false, /*reuse_b=*/false);
  }

  const float bv = bias ? bias[col] : 0.0f;
  const int rbase = row0 + ((lane >> 4) << 3); // C/D: row = v + 8*(lane>=16)
#pragma unroll
  for (int v = 0; v < 8; ++v) {
    const int rr = rbase + v;
    if (rr < M) C[(size_t)rr * HID + col] = acc[v] + bv;
  }
}

// ------------------------- small kernels ------------------------------------

// qr[r,i] = sum_o w[r,i,o]*q[o];  kr[r,i] = sum_o w[r,i,o]*k[o]
__global__ void qk_proj(const float* __restrict__ w, const float* __restrict__ q,
                        const float* __restrict__ k, float* __restrict__ qr,
                        float* __restrict__ kr) {
  const int idx = blockIdx.x * blockDim.x + threadIdx.x;
  if (idx >= NREL * HID) return;
  const float* wr = w + (size_t)idx * HID;     // row (r,i) of w
  float sq = 0.f, sk = 0.f;
#pragma unroll 8
  for (int o = 0; o < HID; ++o) {
    const float ww = wr[o];
    sq += ww * q[o];
    sk += ww * k[o];
  }
  qr[idx] = sq;
  kr[idx] = sk;
}

__global__ void init_node(float* __restrict__ m, float* __restrict__ ssum, int N) {
  const int i = blockIdx.x * blockDim.x + threadIdx.x;
  if (i >= N) return;
  m[i]    = -__builtin_huge_valf();
  ssum[i] = 0.0f;
}

// alpha_e = leaky_relu(h[dst].qr[et] + h[src].kr[et]); segment-max into m[dst]
__global__ void edge_alpha(const int* __restrict__ ei, const int* __restrict__ et,
                           const float* __restrict__ h, const float* __restrict__ qr,
                           const float* __restrict__ kr, float* __restrict__ alpha,
                           float* __restrict__ m, int E) {
  const int e = blockIdx.x * blockDim.x + threadIdx.x;
  if (e >= E) return;
  const int s = ei[e];
  const int d = ei[E + e];
  const int r = et[e];
  const float4* hd = (const float4*)(h + (size_t)d * HID);
  const float4* hs = (const float4*)(h + (size_t)s * HID);
  const float4* qv = (const float4*)(qr + (size_t)r * HID);
  const float4* kv = (const float4*)(kr + (size_t)r * HID);
  float acc = 0.f;
#pragma unroll
  for (int i = 0; i < HID / 4; ++i) {
    const float4 a = hd[i], b = qv[i];
    const float4 c = hs[i], g = kv[i];
    acc += a.x * b.x + a.y * b.y + a.z * b.z + a.w * b.w;
    acc += c.x * g.x + c.y * g.y + c.z * g.z + c.w * g.w;
  }
  const float al = (acc > 0.f) ? acc : NEG_SLOPE * acc;
  alpha[e] = al;
  atomicMaxF(m + d, al);
}

// e_e = exp(alpha - m[dst]); segment-sum into ssum[dst]
__global__ void edge_exp(const int* __restrict__ ei, const float* __restrict__ alpha,
                         const float* __restrict__ m, float* __restrict__ ex,
                         float* __restrict__ ssum, int E) {
  const int e = blockIdx.x * blockDim.x + threadIdx.x;
  if (e >= E) return;
  const int d = ei[E + e];
  const float v = expf(alpha[e] - m[d]);
  ex[e] = v;
  atomicAdd(ssum + d, v);
}

// out[n,c] = bias[c]
__global__ void init_out_bias(float* __restrict__ out, const float* __restrict__ b,
                              int total) {
  const int i = blockIdx.x * blockDim.x + threadIdx.x;
  if (i >= total) return;
  out[i] = b[i & (HID - 1)];
}

// for edges with etype==rel: out[dst] += (ex/max(ssum[dst],eps)) * t[src]
// 4 threads per edge, 16 dims each.
__global__ void edge_scatter(const int* __restrict__ ei, const int* __restrict__ et,
                             const float* __restrict__ ex, const float* __restrict__ ssum,
                             const float* __restrict__ t, float* __restrict__ out,
                             int E, int rel) {
  const int tid  = blockIdx.x * blockDim.x + threadIdx.x;
  const int e    = tid >> 2;
  const int part = tid & 3;
  if (e >= E) return;
  if (et[e] != rel) return;
  const int s = ei[e];
  const int d = ei[E + e];
  const float a = ex[e] / fmaxf(ssum[d], 1e-16f);
  const float* ts = t + (size_t)s * HID + part * 16;
  float* od       = out + (size_t)d * HID + part * 16;
#pragma unroll
  for (int i = 0; i < 16; ++i) atomicAdd(od + i, a * ts[i]);
}

__global__ void relu_copy(const float* __restrict__ in, float* __restrict__ out,
                          int n) {
  const int i = blockIdx.x * blockDim.x + threadIdx.x;
  if (i < n) out[i] = fmaxf(in[i], 0.0f);
}

__global__ void fill0(float* __restrict__ p, int n) {
  const int i = blockIdx.x * blockDim.x + threadIdx.x;
  if (i < n) p[i] = 0.0f;
}

// pooled[batch[n], c] += h[n, c]
__global__ void pool_sum(const int* __restrict__ batch, const float* __restrict__ h,
                         float* __restrict__ pooled, int total) {
  const int i = blockIdx.x * blockDim.x + threadIdx.x;
  if (i >= total) return;
  const int n = i >> 6;
  const int c = i & (HID - 1);
  atomicAdd(pooled + (size_t)batch[n] * HID + c, h[i]);
}

// z[g,o] = relu(sum_k pooled[g,k]*W[k,o] + b[o])   (64x64 @ 64x64, tiny)
__global__ void head_z(const float* __restrict__ pooled, const float* __restrict__ W,
                       const float* __restrict__ b, float* __restrict__ z) {
  const int i = blockIdx.x * blockDim.x + threadIdx.x;
  if (i >= NGRAPH * HID) return;
  const int g = i >> 6, o = i & (HID - 1);
  float acc = b[o];
#pragma unroll 8
  for (int kk = 0; kk < HID; ++kk) acc += pooled[g * HID + kk] * W[kk * HID + o];
  z[i] = fmaxf(acc, 0.0f);
}

// out[g,o] = sum_k z[g,k]*W[k,o] + b[o]   (64x64 @ 64x32, tiny)
__global__ void clf_out(const float* __restrict__ z, const float* __restrict__ W,
                        const float* __restrict__ b, float* __restrict__ out) {
  const int i = blockIdx.x * blockDim.x + threadIdx.x;
  if (i >= NGRAPH * OUTC) return;
  const int g = i >> 5, o = i & (OUTC - 1);
  float acc = b[o];
#pragma unroll 8
  for (int kk = 0; kk < HID; ++kk) acc += z[g * HID + kk] * W[kk * OUTC + o];
  out[i] = acc;
}

// ------------------------- launcher -----------------------------------------

extern "C" void kernel_launch(void* const* d_in, const int* in_sizes, int n_in,
                              void* d_out, int out_size, void* d_ws, size_t ws_size,
                              hipStream_t stream) {
  const float* x    = (const float*)d_in[0];
  const int*   ei   = (const int*)d_in[1];   // [2,E] row-major: src then dst
  const int*   et   = (const int*)d_in[2];
  const int*   batch= (const int*)d_in[3];
  const float* encW = (const float*)d_in[4];
  const float* encB = (const float*)d_in[5];
  const float* w[2] = {(const float*)d_in[6],  (const float*)d_in[10]};
  const float* q[2] = {(const float*)d_in[7],  (const float*)d_in[11]};
  const float* k[2] = {(const float*)d_in[8],  (const float*)d_in[12]};
  const float* b[2] = {(const float*)d_in[9],  (const float*)d_in[13]};
  const float* linW = (const float*)d_in[14];
  const float* linB = (const float*)d_in[15];
  const float* clfW = (const float*)d_in[16];
  const float* clfB = (const float*)d_in[17];

  const int N   = in_sizes[3];               // nodes (batch array length)
  const int E   = in_sizes[2];               // edges (edge_type length)
  const int INC = in_sizes[4] / HID;         // encoder input channels (128)

  // workspace layout (floats) — ~45.2 MB total
  float* ws     = (float*)d_ws;
  float* hA     = ws;                         // [N,64] current layer input
  float* tB     = hA     + (size_t)N * HID;   // [N,64] per-relation transform
  float* outB   = tB     + (size_t)N * HID;   // [N,64] layer output accumulator
  float* alpha  = outB   + (size_t)N * HID;   // [E]
  float* ex     = alpha  + E;                 // [E]
  float* m      = ex     + E;                 // [N] segment max
  float* ssum   = m      + N;                 // [N] segment sum
  float* qr     = ssum   + N;                 // [8,64]
  float* kr     = qr     + NREL * HID;        // [8,64]
  float* pooled = kr     + NREL * HID;        // [64,64]
  float* z      = pooled + NGRAPH * HID;      // [64,64]

  const int gemmGrid = (N + 15) / 16;
  const int nodeGrid = (N + 255) / 256;
  const int nhGrid   = (N * HID + 255) / 256;
  const int edgeGrid = (E + 255) / 256;
  const int scatGrid = (E * 4 + 255) / 256;

  // encoder: h = x @ enc_W + enc_b
  wmma_gemm_n64<<<gemmGrid, 128, 0, stream>>>(x, encW, encB, hA, N, INC);

  for (int L = 0; L < 2; ++L) {
    qk_proj<<<(NREL * HID + 255) / 256, 256, 0, stream>>>(w[L], q[L], k[L], qr, kr);
    init_node<<<nodeGrid, 256, 0, stream>>>(m, ssum, N);
    edge_alpha<<<edgeGrid, 256, 0, stream>>>(ei, et, hA, qr, kr, alpha, m, E);
    edge_exp<<<edgeGrid, 256, 0, stream>>>(ei, alpha, m, ex, ssum, E);
    init_out_bias<<<nhGrid, 256, 0, stream>>>(outB, b[L], N * HID);
    for (int r = 0; r < NREL; ++r) {
      // t = h @ w[r]  (WMMA), then scatter a_e * t[src] into out[dst]
      wmma_gemm_n64<<<gemmGrid, 128, 0, stream>>>(
          hA, w[L] + (size_t)r * HID * HID, nullptr, tB, N, HID);
      edge_scatter<<<scatGrid, 256, 0, stream>>>(ei, et, ex, ssum, tB, outB, E, r);
    }
    if (L == 0)  // inter-layer ReLU; layer-1 output is pooled un-activated
      relu_copy<<<nhGrid, 256, 0, stream>>>(outB, hA, N * HID);
  }

  fill0<<<(NGRAPH * HID + 255) / 256, 256, 0, stream>>>(pooled, NGRAPH * HID);
  pool_sum<<<nhGrid, 256, 0, stream>>>(batch, outB, pooled, N * HID);
  head_z<<<(NGRAPH * HID + 255) / 256, 256, 0, stream>>>(pooled, linW, linB, z);
  clf_out<<<(NGRAPH * OUTC + 255) / 256, 256, 0, stream>>>(z, clfW, clfB,
                                                           (float*)d_out);
}